// DeepRNN_17102559773047
// MI455X (gfx1250) — compile-verified
//
#include <hip/hip_runtime.h>
#include <stdint.h>

// CDNA5 / gfx1250. Wave32. WMMA f32 16x16x4 + async global->LDS staging.

typedef __attribute__((ext_vector_type(2))) float v2f;
typedef __attribute__((ext_vector_type(8))) float v8f;

#define VOCAB  32000
#define HIDDEN 512
#define LAYERS 3
#define BATCH  32
#define SEQ    64

#define BH      (BATCH * HIDDEN)      // 16384 floats per layer-state
#define STSZ    (LAYERS * BH)         // 49152 floats (one parity buffer)
#define HROWS   (SEQ * BATCH)         // 2048 rows of the stacked H matrix
#define OUT_OFF ((size_t)HROWS * VOCAB)

static __device__ __forceinline__ v8f wmma_f32(v2f a, v2f b, v8f c) {
  // 8 args: (neg_a, A, neg_b, B, c_mod, C, reuse_a, reuse_b)
  return __builtin_amdgcn_wmma_f32_16x16x4_f32(false, a, false, b, (short)0, c,
                                               false, false);
}

// Async 16-byte global -> LDS copy (per lane). Tracked by ASYNCcnt.
// lds = low 32 bits of the generic shared pointer == LDS byte address.
static __device__ __forceinline__ void async_b128(uint32_t lds, const float* g) {
  asm volatile("global_load_async_to_lds_b128 %0, %1, off"
               :: "v"(lds), "v"((uint64_t)(uintptr_t)g)
               : "memory");
}

// ---------------------------------------------------------------------------
// Kernel 1: serial recurrence. One persistent workgroup (32 waves on one WGP).
// Each wave owns a 16-wide column slice of the 512-wide hidden state and
// computes two 16x16 WMMA tiles (rows 0-15 and 16-31 of the batch).
// Ping-pong state buffers in d_ws; workgroup barrier between layer phases.
// ---------------------------------------------------------------------------
__global__ __launch_bounds__(1024, 1)
void rnn_recurrence(const int*   __restrict__ x,       // (BATCH, SEQ)
                    const float* __restrict__ st_in,   // (LAYERS, BATCH, HIDDEN)
                    const float* __restrict__ W_xi,    // (VOCAB, HIDDEN)
                    const float* __restrict__ W_xh,    // (LAYERS-1, HIDDEN, HIDDEN)
                    const float* __restrict__ W_hh,    // (LAYERS, HIDDEN, HIDDEN)
                    const float* __restrict__ b_h,     // (LAYERS, HIDDEN)
                    float*       __restrict__ ws_H,    // (HROWS, HIDDEN) out
                    float*       __restrict__ ws_st,   // 2 * STSZ scratch
                    float*       __restrict__ out_state) {
  const int tid   = threadIdx.x;
  const int wv    = tid >> 5;        // wave id 0..31 -> column slice
  const int lane  = tid & 31;
  const int m     = lane & 15;       // A-row / B-col within tile
  const int hi    = lane >> 4;       // 0: lanes 0-15, 1: lanes 16-31
  const int koff  = hi << 1;         // A/B fragments: upper half carries K+2,K+3
  const int rbase = hi << 3;         // C fragment: upper half carries M+8
  const int nn    = (wv << 4) + m;   // this lane's hidden column 0..511

  // init parity-0 state buffer from the input state
  for (int i = tid; i < STSZ; i += 1024) ws_st[i] = st_in[i];
  __syncthreads();

  for (int t = 0; t < SEQ; ++t) {
    const float* stp = ws_st + (t & 1) * STSZ;         // previous-step states
    float*       stn = ws_st + ((t & 1) ^ 1) * STSZ;   // this-step states

    for (int l = 0; l < LAYERS; ++l) {
      v8f c0 = {};  // rows 0-15
      v8f c1 = {};  // rows 16-31

      // recurrent part: st[l] @ W_hh[l]
      {
        const float* Ah = stp + l * BH;
        const float* Wh = W_hh + (size_t)l * HIDDEN * HIDDEN;
        for (int k = 0; k < HIDDEN; k += 4) {
          const int ka = k + koff;
          v2f a0, a1, b;
          a0.x = Ah[m * HIDDEN + ka];          a0.y = Ah[m * HIDDEN + ka + 1];
          a1.x = Ah[(m + 16) * HIDDEN + ka];   a1.y = Ah[(m + 16) * HIDDEN + ka + 1];
          b.x  = Wh[ka * HIDDEN + nn];         b.y  = Wh[(ka + 1) * HIDDEN + nn];
          c0 = wmma_f32(a0, b, c0);
          c1 = wmma_f32(a1, b, c1);
        }
      }
      // feed-forward part (layers 1,2): h_{l-1}(t) @ W_xh[l-1]
      if (l > 0) {
        const float* Ax = stn + (l - 1) * BH;  // written last phase, barrier'd
        const float* Wx = W_xh + (size_t)(l - 1) * HIDDEN * HIDDEN;
        for (int k = 0; k < HIDDEN; k += 4) {
          const int ka = k + koff;
          v2f a0, a1, b;
          a0.x = Ax[m * HIDDEN + ka];          a0.y = Ax[m * HIDDEN + ka + 1];
          a1.x = Ax[(m + 16) * HIDDEN + ka];   a1.y = Ax[(m + 16) * HIDDEN + ka + 1];
          b.x  = Wx[ka * HIDDEN + nn];         b.y  = Wx[(ka + 1) * HIDDEN + nn];
          c0 = wmma_f32(a0, b, c0);
          c1 = wmma_f32(a1, b, c1);
        }
      }

      // epilogue: (+ embedding row for layer 0) + bias, tanh, store
      const float bh = b_h[l * HIDDEN + nn];
      #pragma unroll
      for (int r = 0; r < 8; ++r) {
        const int r0 = rbase + r;  // row within 0..15 block of c0
        float v0 = c0[r] + bh;
        float v1 = c1[r] + bh;
        if (l == 0) {
          // one_hot(x_t) @ W_xi == W_xi[x_t]  (x is (batch, seq) row-major)
          v0 += W_xi[(size_t)x[r0 * SEQ + t] * HIDDEN + nn];
          v1 += W_xi[(size_t)x[(r0 + 16) * SEQ + t] * HIDDEN + nn];
        }
        v0 = tanhf(v0);
        v1 = tanhf(v1);
        stn[l * BH + r0 * HIDDEN + nn]        = v0;
        stn[l * BH + (r0 + 16) * HIDDEN + nn] = v1;
        if (l == LAYERS - 1) {
          ws_H[(size_t)(t * BATCH + r0) * HIDDEN + nn]      = v0;
          ws_H[(size_t)(t * BATCH + r0 + 16) * HIDDEN + nn] = v1;
        }
      }
      __syncthreads();
    }
  }

  // after t=63 the newest state sits in the parity-0 buffer
  for (int i = tid; i < STSZ; i += 1024) out_state[i] = ws_st[i];
}

// ---------------------------------------------------------------------------
// Kernel 2: output projection  C(2048x32000) = H(2048x512) @ W_ho + b_o.
// Block = 8 waves, 64x256 C tile. Double-buffered LDS staging of A(64xKC)
// and B(KCx256) per K-chunk via global_load_async_to_lds_b128 (ASYNCcnt),
// overlapping the L2->LDS stream with WMMA on the previous chunk.
// ---------------------------------------------------------------------------
#define KC 16                        // K-chunk depth
#define NKC (HIDDEN / KC)            // 32 chunks

__global__ __launch_bounds__(256, 2)
void out_gemm(const float* __restrict__ H,     // (HROWS, HIDDEN)
              const float* __restrict__ W_ho,  // (HIDDEN, VOCAB)
              const float* __restrict__ b_o,   // (VOCAB)
              float*       __restrict__ out) { // (HROWS, VOCAB)
  __shared__ float As[2][64][KC];    //  8 KB: block's 64 H rows, KC-deep
  __shared__ float Bs[2][KC][256];   // 32 KB: KC W_ho rows, block's 256 cols

  const int tid   = threadIdx.x;
  const int wv    = tid >> 5;
  const int lane  = tid & 31;
  const int m     = lane & 15;
  const int hi    = lane >> 4;
  const int koff  = hi << 1;
  const int rbase = hi << 3;
  const int mpair = wv >> 2;  // 0..1 : which 32-row half of the block tile
  const int ngrp  = wv & 3;   // 0..3 : which 64-col quarter of the block tile

  const int rowblk0 = blockIdx.y << 6;   // first of 64 C rows
  const int ncol0   = blockIdx.x << 8;   // first of 256 C cols

  const int rowloc0 = (mpair << 5) + m;  // A row (local) for M-tile 0
  const int rowloc1 = rowloc0 + 16;      // M-tile 1
  int colloc[4];
  #pragma unroll
  for (int j = 0; j < 4; ++j) colloc[j] = ((ngrp << 2) + j) * 16 + m;

  // per-chunk async issue: 1 A unit + 4 B units per thread (5 per wave-instr)
  // A: 64 rows x KC floats  = 256 b128 units  (1/thread)
  // B: KC rows x 256 floats = 1024 b128 units (4/thread)
  const int a_row = tid >> 2;
  const int a_cu  = (tid & 3) << 2;

  {
    const int k0 = 0;
    async_b128((uint32_t)(uintptr_t)&As[0][a_row][a_cu],
               H + (size_t)(rowblk0 + a_row) * HIDDEN + k0 + a_cu);
    #pragma unroll
    for (int i = 0; i < 4; ++i) {
      const int u  = (i << 8) + tid;
      const int kk = u >> 6;
      const int cu = (u & 63) << 2;
      async_b128((uint32_t)(uintptr_t)&Bs[0][kk][cu],
                 W_ho + (size_t)(k0 + kk) * VOCAB + ncol0 + cu);
    }
  }

  v8f acc[2][4];
  #pragma unroll
  for (int i = 0; i < 2; ++i)
    #pragma unroll
    for (int j = 0; j < 4; ++j) acc[i][j] = (v8f){};

  for (int c = 0; c < NKC; ++c) {
    const int buf = c & 1;
    if (c + 1 < NKC) {
      // keep next chunk in flight while we compute on this one
      const int k0 = (c + 1) * KC;
      async_b128((uint32_t)(uintptr_t)&As[buf ^ 1][a_row][a_cu],
                 H + (size_t)(rowblk0 + a_row) * HIDDEN + k0 + a_cu);
      #pragma unroll
      for (int i = 0; i < 4; ++i) {
        const int u  = (i << 8) + tid;
        const int kk = u >> 6;
        const int cu = (u & 63) << 2;
        async_b128((uint32_t)(uintptr_t)&Bs[buf ^ 1][kk][cu],
                   W_ho + (size_t)(k0 + kk) * VOCAB + ncol0 + cu);
      }
      asm volatile("s_wait_asynccnt 0x5" ::: "memory");  // chunk c landed
    } else {
      asm volatile("s_wait_asynccnt 0x0" ::: "memory");
    }
    __syncthreads();

    #pragma unroll
    for (int kl = 0; kl < KC; kl += 4) {
      const int kk = kl + koff;
      v2f a0 = *(const v2f*)&As[buf][rowloc0][kk];
      v2f a1 = *(const v2f*)&As[buf][rowloc1][kk];
      #pragma unroll
      for (int j = 0; j < 4; ++j) {
        v2f b;
        b.x = Bs[buf][kk][colloc[j]];
        b.y = Bs[buf][kk + 1][colloc[j]];
        acc[0][j] = wmma_f32(a0, b, acc[0][j]);
        acc[1][j] = wmma_f32(a1, b, acc[1][j]);
      }
    }
    __syncthreads();  // chunk buf may be overwritten by next issue
  }

  #pragma unroll
  for (int i = 0; i < 2; ++i) {
    const int mrow = rowblk0 + (mpair << 5) + (i << 4) + rbase;
    #pragma unroll
    for (int j = 0; j < 4; ++j) {
      const int col = ncol0 + colloc[j];
      const float bo = b_o[col];
      #pragma unroll
      for (int r = 0; r < 8; ++r) {
        out[(size_t)(mrow + r) * VOCAB + col] = acc[i][j][r] + bo;
      }
    }
  }
}

// ---------------------------------------------------------------------------
extern "C" void kernel_launch(void* const* d_in, const int* in_sizes, int n_in,
                              void* d_out, int out_size, void* d_ws, size_t ws_size,
                              hipStream_t stream) {
  (void)in_sizes; (void)n_in; (void)out_size; (void)ws_size;
  const int*   x     = (const int*)  d_in[0];
  const float* state = (const float*)d_in[1];
  const float* W_xi  = (const float*)d_in[2];
  const float* W_xh  = (const float*)d_in[3];
  const float* W_hh  = (const float*)d_in[4];
  const float* b_h   = (const float*)d_in[5];
  const float* W_ho  = (const float*)d_in[6];
  const float* b_o   = (const float*)d_in[7];

  float* out   = (float*)d_out;
  float* ws    = (float*)d_ws;
  float* ws_H  = ws;                                   // 2048*512 floats
  float* ws_st = ws + (size_t)HROWS * HIDDEN;          // 2 * STSZ floats

  // 1) serial recurrence on one WGP: produces H (all h2(t)) + final state
  rnn_recurrence<<<1, 1024, 0, stream>>>(x, state, W_xi, W_xh, W_hh, b_h,
                                         ws_H, ws_st, out + OUT_OFF);

  // 2) big output projection: 2048 x 32000 x 512 GEMM (exact tiling, no guards)
  dim3 grid(VOCAB / 256, HROWS / 64);
  out_gemm<<<grid, 256, 0, stream>>>(ws_H, W_ho, b_o, out);
}